// DotProductAttention_85426899517567
// MI455X (gfx1250) — compile-verified
//
#include <hip/hip_runtime.h>

typedef float v2f __attribute__((ext_vector_type(2)));
typedef float v4f __attribute__((ext_vector_type(4)));
typedef float v8f __attribute__((ext_vector_type(8)));

#define LQ    4096
#define BATCH 4
#define DHEAD 64
#define NEGV  (-1e9f)
#define LDP   66   // padded LDS row stride (floats): kills 16-way bank conflicts, keeps 8B align

// Load one 64x64 f32 block (rows blk*64..+64 of a [L,B,D] tensor, batch b) into 8 float4 regs.
__device__ __forceinline__ void load_block_regs(float4 (&reg)[8], const float* __restrict__ src,
                                                int blk, int b, int tid) {
#pragma unroll
    for (int j = 0; j < 8; ++j) {
        int i  = tid + j * 128;
        int r  = i >> 4, c4 = i & 15;
        reg[j] = ((const float4*)(src + ((size_t)(blk * 64 + r) * BATCH + b) * DHEAD))[c4];
    }
}

// Spill the 8 float4 regs into a padded 64xLDP LDS block.
__device__ __forceinline__ void store_block_lds(float* __restrict__ lds, const float4 (&reg)[8],
                                                int tid, float scale) {
#pragma unroll
    for (int j = 0; j < 8; ++j) {
        int i = tid + j * 128;
        int r = i >> 4, c4 = i & 15;
        float* dst = &lds[r * LDP + c4 * 4];
        dst[0] = reg[j].x * scale; dst[1] = reg[j].y * scale;
        dst[2] = reg[j].z * scale; dst[3] = reg[j].w * scale;
    }
}

__launch_bounds__(128)
__global__ void DotProductAttention_85426899517567_kernel(
    const float* __restrict__ q,
    const float* __restrict__ k,
    const float* __restrict__ v,
    const unsigned char* __restrict__ mask,   // [B, L] bool (1 byte)
    float* __restrict__ out,                  // [L, B, D]
    float* __restrict__ attn)                 // [B, L, L]
{
    __shared__ float qs[64 * LDP];   // Q block (scaled)
    __shared__ float ks[64 * LDP];   // K block (pass 1) / P staging (pass 2)
    __shared__ float vs[64 * LDP];   // V block (pass 2)
    __shared__ float rowmax_s[64];
    __shared__ float rowinv_s[64];

    const int b    = blockIdx.x >> 6;       // batch
    const int qb   = blockIdx.x & 63;       // query block of 64 rows
    const int tid  = threadIdx.x;
    const int wave = tid >> 5;              // 0..3, wave owns q rows [wave*16, +16)
    const int lane = tid & 31;
    const int half = lane >> 4;             // lane half selects K-pair / row+8
    const int ln   = lane & 15;
    const int wrow = wave * 16;

    float* attn_b = attn + (size_t)b * LQ * LQ + (size_t)(qb * 64) * LQ;

    // ---- load Q block (64 rows x 64) into LDS, pre-scaled by D^-0.5 = 0.125 ----
    {
        float4 qreg[8];
        load_block_regs(qreg, q, qb, b, tid);
        store_block_lds(qs, qreg, tid, 0.125f);
    }

    float rmax[8];
#pragma unroll
    for (int i = 0; i < 8; ++i) rmax[i] = -__builtin_inff();

    // ================= PASS 1: S = Q K^T, mask, store raw S, track row max =================
    {
        float4 kreg[8];
        load_block_regs(kreg, k, 0, b, tid);           // preload kb=0
        for (int kb = 0; kb < 64; ++kb) {
            __syncthreads();                            // prior compute done reading ks
            store_block_lds(ks, kreg, tid, 1.0f);
            if (kb < 63) load_block_regs(kreg, k, kb + 1, b, tid);   // overlap with compute
            __syncthreads();                            // ks ready

            for (int nt = 0; nt < 4; ++nt) {            // 4 key tiles of 16
                v8f c = {0.f, 0.f, 0.f, 0.f, 0.f, 0.f, 0.f, 0.f};
#pragma unroll
                for (int kc = 0; kc < 16; ++kc) {       // D=64 in chunks of 4
                    int kk = kc * 4 + half * 2;
                    v2f a  = *(const v2f*)&qs[(wrow + ln) * LDP + kk];     // A: Q[m][kk..kk+1]
                    v2f bb = *(const v2f*)&ks[(nt * 16 + ln) * LDP + kk];  // B: K^T[kk][n] = K[n][kk]
                    c = __builtin_amdgcn_wmma_f32_16x16x4_f32(false, a, false, bb,
                                                              (short)0, c, false, false);
                }
                int col = kb * 64 + nt * 16 + ln;
                bool msk = mask[b * LQ + col] != 0;
#pragma unroll
                for (int i = 0; i < 8; ++i) {
                    float s = msk ? NEGV : c[i];
                    attn_b[(size_t)(wrow + i + half * 8) * LQ + col] = s;
                    rmax[i] = fmaxf(rmax[i], s);
                }
            }
        }
    }

    // reduce row max across the 16 lanes of each half
#pragma unroll
    for (int i = 0; i < 8; ++i) {
        float m = rmax[i];
        m = fmaxf(m, __shfl_xor(m, 1, 32));
        m = fmaxf(m, __shfl_xor(m, 2, 32));
        m = fmaxf(m, __shfl_xor(m, 4, 32));
        m = fmaxf(m, __shfl_xor(m, 8, 32));
        rmax[i] = m;
    }

    // ================= PASS 2: O = exp(S - max) @ V, accumulate row sums =================
    float rsum[8];
#pragma unroll
    for (int i = 0; i < 8; ++i) rsum[i] = 0.f;
    v8f oacc[4];
#pragma unroll
    for (int nt = 0; nt < 4; ++nt) oacc[nt] = (v8f){0.f,0.f,0.f,0.f,0.f,0.f,0.f,0.f};

    {
        float4 vreg[8];
        load_block_regs(vreg, v, 0, b, tid);           // preload kb=0
        for (int kb = 0; kb < 64; ++kb) {
            __syncthreads();                            // prior compute done reading vs/ks
            store_block_lds(vs, vreg, tid, 1.0f);
            if (kb < 63) load_block_regs(vreg, v, kb + 1, b, tid);   // overlap with compute

            // reload raw S (L2-hot), compute P, stage into wave-private LDS region
            for (int nt = 0; nt < 4; ++nt) {
                int col = kb * 64 + nt * 16 + ln;
#pragma unroll
                for (int i = 0; i < 8; ++i) {
                    int rl = i + half * 8;
                    float s = attn_b[(size_t)(wrow + rl) * LQ + col];
                    float p = __expf(s - rmax[i]);
                    rsum[i] += p;
                    ks[(wrow + rl) * LDP + nt * 16 + ln] = p;
                }
            }
            __syncthreads();                            // vs + P staging ready

            for (int nt = 0; nt < 4; ++nt) {            // 4 D tiles of 16
#pragma unroll
                for (int kc = 0; kc < 16; ++kc) {       // 64 keys in chunks of 4
                    int kk = kc * 4 + half * 2;
                    v2f a = *(const v2f*)&ks[(wrow + ln) * LDP + kk];    // A: P[m][kk..kk+1]
                    v2f bb;
                    bb.x = vs[kk * LDP + nt * 16 + ln];                  // B: V[kk][d]
                    bb.y = vs[(kk + 1) * LDP + nt * 16 + ln];
                    oacc[nt] = __builtin_amdgcn_wmma_f32_16x16x4_f32(false, a, false, bb,
                                                                     (short)0, oacc[nt], false, false);
                }
            }
        }
    }

    // reduce row sums, compute reciprocals, stash per-row stats for pass 3
    float rinv[8];
#pragma unroll
    for (int i = 0; i < 8; ++i) {
        float s = rsum[i];
        s += __shfl_xor(s, 1, 32);
        s += __shfl_xor(s, 2, 32);
        s += __shfl_xor(s, 4, 32);
        s += __shfl_xor(s, 8, 32);
        rinv[i] = 1.0f / s;
    }
    if (ln == 0) {
#pragma unroll
        for (int i = 0; i < 8; ++i) {
            rowmax_s[wrow + i + half * 8] = rmax[i];
            rowinv_s[wrow + i + half * 8] = rinv[i];
        }
    }

    // write O / rowsum to out in [L, B, D] layout
    for (int nt = 0; nt < 4; ++nt) {
#pragma unroll
        for (int i = 0; i < 8; ++i) {
            int m = wrow + i + half * 8;
            int d = nt * 16 + ln;
            out[((size_t)(qb * 64 + m) * BATCH + b) * DHEAD + d] = oacc[nt][i] * rinv[i];
        }
    }

    __syncthreads();

    // ========== PASS 3: normalize attn block: exp(s - max) * inv, coalesced v4f ==========
    // Raw S is last-use here and the result is never re-read -> non-temporal load + store
    // keep the 192MB L2 free for other WGs' still-live raw-S slabs.
    for (int f = tid; f < 64 * 1024; f += 128) {   // 64 rows * 1024 float4 per row
        int r  = f >> 10;
        int c4 = f & 1023;
        float mrow = rowmax_s[r];
        float inv  = rowinv_s[r];
        v4f* p4 = (v4f*)(attn_b + (size_t)r * LQ) + c4;
        v4f sv = __builtin_nontemporal_load(p4);
        sv.x = __expf(sv.x - mrow) * inv;
        sv.y = __expf(sv.y - mrow) * inv;
        sv.z = __expf(sv.z - mrow) * inv;
        sv.w = __expf(sv.w - mrow) * inv;
        __builtin_nontemporal_store(sv, p4);
    }
}

extern "C" void kernel_launch(void* const* d_in, const int* in_sizes, int n_in,
                              void* d_out, int out_size, void* d_ws, size_t ws_size,
                              hipStream_t stream) {
    const float* q = (const float*)d_in[0];
    const float* k = (const float*)d_in[1];
    const float* v = (const float*)d_in[2];
    const unsigned char* mask = (const unsigned char*)d_in[3];
    float* out  = (float*)d_out;                       // [L, B, D] = 1,048,576 floats
    float* attn = out + (size_t)LQ * BATCH * DHEAD;    // [B, L, L] = 67,108,864 floats

    dim3 grid(BATCH * (LQ / 64));   // 256 workgroups: (batch, 64-row query block)
    dim3 block(128);                // 4 waves of 32
    hipLaunchKernelGGL(DotProductAttention_85426899517567_kernel,
                       grid, block, 0, stream, q, k, v, mask, out, attn);
}